// InferModule_28260884808445
// MI455X (gfx1250) — compile-verified
//
#include <hip/hip_runtime.h>

#define GAMMA_F 0.01f
#define INVG_F  100.0f

constexpr int B = 16, G = 8192, C = 16, NSTEP = 4;
constexpr int BPB     = 4;     // batch rows staged per block (4 x 32KB = 128KB LDS)
constexpr int BLOCK_G = 128;   // g-columns per block in pass A
constexpr int THREADS = 256;   // 8 waves (wave32)

typedef __attribute__((ext_vector_type(2))) float v2f;
typedef __attribute__((ext_vector_type(8))) float v8f;

// One (c,g) index cell: 24 contiguous ints (96B, 16B aligned) of I(C,G,S=8,L=3)
struct Cell { int4 a, b, c, d, e, f; };

__device__ __forceinline__ Cell load_cell(const int* __restrict__ I, int c, int g) {
  const int4* ip = reinterpret_cast<const int4*>(I + (long)(c * G + g) * 24);
  Cell k;
  k.a = ip[0]; k.b = ip[1]; k.c = ip[2]; k.d = ip[3]; k.e = ip[4]; k.f = ip[5];
  return k;
}

// Stable gamma-logsumexp over S=8 of products over L=3, gathered from one LDS row.
__device__ __forceinline__ float ls1_eval(const float* __restrict__ Rb, const Cell& k) {
  float p0 = Rb[k.a.x] * Rb[k.a.y] * Rb[k.a.z];
  float p1 = Rb[k.a.w] * Rb[k.b.x] * Rb[k.b.y];
  float p2 = Rb[k.b.z] * Rb[k.b.w] * Rb[k.c.x];
  float p3 = Rb[k.c.y] * Rb[k.c.z] * Rb[k.c.w];
  float p4 = Rb[k.d.x] * Rb[k.d.y] * Rb[k.d.z];
  float p5 = Rb[k.d.w] * Rb[k.e.x] * Rb[k.e.y];
  float p6 = Rb[k.e.z] * Rb[k.e.w] * Rb[k.f.x];
  float p7 = Rb[k.f.y] * Rb[k.f.z] * Rb[k.f.w];
  float m = fmaxf(fmaxf(fmaxf(p0, p1), fmaxf(p2, p3)),
                  fmaxf(fmaxf(p4, p5), fmaxf(p6, p7)));
  float s = __expf((p0 - m) * INVG_F) + __expf((p1 - m) * INVG_F) +
            __expf((p2 - m) * INVG_F) + __expf((p3 - m) * INVG_F) +
            __expf((p4 - m) * INVG_F) + __expf((p5 - m) * INVG_F) +
            __expf((p6 - m) * INVG_F) + __expf((p7 - m) * INVG_F);
  return m + GAMMA_F * __logf(s);
}

__device__ __forceinline__ float pick8(const v8f& a, int comp) {
  return (comp == 0) ? a[0] : (comp == 1) ? a[1] :
         (comp == 2) ? a[2] : (comp == 3) ? a[3] :
         (comp == 4) ? a[4] : (comp == 5) ? a[5] :
         (comp == 6) ? a[6] : a[7];
}

__global__ void init_max_kernel(int* max1, int* max3) {
  max1[0] = __float_as_int(1.0f);   // fold max(.,1.0) into the atomic init
  max3[0] = __float_as_int(1.0f);
}

// Pass A: stage BPB=4 batch rows (128KB) in LDS; each wave owns a 16-g tile.
// Per K-chunk, the 2x24 index cell is loaded ONCE and reused for all 4 batches
// (4x fewer scattered VMEM fetches). ls1 values are produced directly in the
// V_WMMA_F32_16X16X4_F32 A-layout and contracted against the softmax(W)
// broadcast with 4 independent accumulators (one per batch).
__global__ void __launch_bounds__(THREADS)
passA_kernel(const float* __restrict__ R, const float* __restrict__ W,
             const int* __restrict__ I, float* __restrict__ U,
             int* __restrict__ max1) {
  __shared__ float ldsR[BPB * G];   // 128 KB (<= 320KB/WGP on CDNA5)
  __shared__ float ldsW[C];
  const int b0  = blockIdx.y * BPB;
  const int tid = threadIdx.x;

  // Stage 4 contiguous R rows into LDS as float4
  const float4* src = reinterpret_cast<const float4*>(R + (size_t)b0 * G);
  float4* dst = reinterpret_cast<float4*>(ldsR);
#pragma unroll
  for (int i = 0; i < (BPB * G / 4) / THREADS; ++i)
    dst[tid + i * THREADS] = src[tid + i * THREADS];

  if (tid == 0) {  // softmax over the 16 clause weights (m=1)
    float wv[C], wmax = -1e30f;
    for (int c = 0; c < C; ++c) { wv[c] = W[c]; wmax = fmaxf(wmax, wv[c]); }
    float ssum = 0.f;
    for (int c = 0; c < C; ++c) { wv[c] = __expf(wv[c] - wmax); ssum += wv[c]; }
    float inv = 1.0f / ssum;
    for (int c = 0; c < C; ++c) ldsW[c] = wv[c] * inv;
  }
  __syncthreads();

  const int wave = tid >> 5, lane = tid & 31;
  const int half = lane >> 4;        // A-layout: lanes 0-15 hold K base, 16-31 hold K+2
  const int gl   = lane & 15;        // A-layout: M = lane & 15 (our M == g within tile)
  const int g0   = blockIdx.x * BLOCK_G + wave * 16;
  const int g    = g0 + gl;

  v8f acc[BPB];
#pragma unroll
  for (int bb = 0; bb < BPB; ++bb) acc[bb] = (v8f){};
  float lmax = 0.0f;

#pragma unroll
  for (int q = 0; q < 4; ++q) {      // K = c, contracted 4 at a time
    const int c0 = 4 * q + 2 * half;
    if (q < 3)  // prefetch next chunk's index cell (global_prefetch_b8)
      __builtin_prefetch(I + (long)((c0 + 4) * G + g) * 24, 0, 0);
    const Cell k0 = load_cell(I, c0,     g);   // -> A vgpr0 (K = c0)
    const Cell k1 = load_cell(I, c0 + 1, g);   // -> A vgpr1 (K = c0+1)
    v2f Bv; Bv.x = ldsW[c0]; Bv.y = ldsW[c0 + 1];   // row-constant B: all N equal
#pragma unroll
    for (int bb = 0; bb < BPB; ++bb) {         // reuse indices across 4 batches
      const float* Rb = ldsR + bb * G;
      float l0 = ls1_eval(Rb, k0);
      float l1 = ls1_eval(Rb, k1);
      lmax = fmaxf(lmax, fmaxf(l0, l1));
      v2f Av; Av.x = l0; Av.y = l1;
      acc[bb] = __builtin_amdgcn_wmma_f32_16x16x4_f32(
          false, Av, false, Bv, (short)0, acc[bb], false, false);
    }
  }

  // D layout: vgpr r holds row M=r (lanes 0-15) / M=8+r (lanes 16-31) at N=lane&15.
  // All 16 columns identical; diagonal lanes write U.
  const bool writer = (lane < 8) || (lane >= 16 && lane < 24);
  const int  comp   = (lane < 8) ? lane : (lane - 16);
  const int  mrow   = (lane < 8) ? lane : (lane - 8);
  if (writer) {
#pragma unroll
    for (int bb = 0; bb < BPB; ++bb)
      U[(size_t)(b0 + bb) * G + g0 + mrow] = pick8(acc[bb], comp);
  }

  // wave32 max-reduce of ls1, one atomic per wave (nonneg float == int order)
#pragma unroll
  for (int off = 16; off > 0; off >>= 1)
    lmax = fmaxf(lmax, __shfl_xor(lmax, off, 32));
  if (lane == 0) atomicMax(max1, __float_as_int(lmax));
}

// Pass B: r_step = U / M1 (second softor normalization provably = 1);
// ls3 = stable gamma-logsumexp of {R, r_step}; track global max(ls3).
__global__ void __launch_bounds__(256)
passB_kernel(const float* __restrict__ R, const float* __restrict__ U,
             const int* __restrict__ max1, float* __restrict__ ls3buf,
             int* __restrict__ max3) {
  const int t = blockIdx.x * 256 + threadIdx.x;
  const float M1 = __int_as_float(max1[0]);
  const float u  = U[t] / M1;
  const float r  = R[t];
  const float mx = fmaxf(r, u), mn = fminf(r, u);
  const float ls3 = mx + GAMMA_F * __logf(1.0f + __expf((mn - mx) * INVG_F));
  ls3buf[t] = ls3;
  float v = ls3;
#pragma unroll
  for (int off = 16; off > 0; off >>= 1)
    v = fmaxf(v, __shfl_xor(v, off, 32));
  if ((threadIdx.x & 31) == 0) atomicMax(max3, __float_as_int(v));
}

// Pass C: normalize -> next R
__global__ void __launch_bounds__(256)
passC_kernel(const float* __restrict__ ls3buf, const int* __restrict__ max3,
             float* __restrict__ Rout) {
  const int t = blockIdx.x * 256 + threadIdx.x;
  Rout[t] = ls3buf[t] / __int_as_float(max3[0]);
}

extern "C" void kernel_launch(void* const* d_in, const int* in_sizes, int n_in,
                              void* d_out, int out_size, void* d_ws, size_t ws_size,
                              hipStream_t stream) {
  const float* x = (const float*)d_in[0];
  const float* W = (const float*)d_in[1];
  const int*   I = (const int*)d_in[2];
  float* out = (float*)d_out;

  char* ws = (char*)d_ws;
  int*   max1 = (int*)ws;
  int*   max3 = (int*)(ws + 64);
  float* U    = (float*)(ws + 256);
  float* LS3  = (float*)(ws + 256 + sizeof(float) * (size_t)B * G);

  const dim3 gridA(G / BLOCK_G, B / BPB);   // 64 x 4 = 256 blocks
  const int n = B * G;

  for (int step = 0; step < NSTEP; ++step) {
    const float* Rsrc = (step == 0) ? x : out;
    init_max_kernel<<<1, 1, 0, stream>>>(max1, max3);
    passA_kernel<<<gridA, THREADS, 0, stream>>>(Rsrc, W, I, U, max1);
    passB_kernel<<<n / 256, 256, 0, stream>>>(Rsrc, U, max1, LS3, max3);
    passC_kernel<<<n / 256, 256, 0, stream>>>(LS3, max3, out);
  }
}